// SpatialNCA_49581102465544
// MI455X (gfx1250) — compile-verified
//
#include <hip/hip_runtime.h>

typedef float v2f __attribute__((ext_vector_type(2)));
typedef float v8f __attribute__((ext_vector_type(8)));

#define DD 64

__device__ __forceinline__ v8f wmma4(v2f a, v2f b, v8f c) {
    // V_WMMA_F32_16X16X4_F32 : D = A(16x4 f32) x B(4x16 f32) + C(16x16 f32)
    return __builtin_amdgcn_wmma_f32_16x16x4_f32(false, a, false, b, (short)0, c, false, false);
}

// Swizzled weight layout: Wz[(k>>1)*128 + 2*n + (k&1)] = W[k*64 + n]
// -> lane's B fragment (W[kk][n], W[kk+1][n]) is one aligned ds_load_b64.

// ---------------------------------------------------------------------------
// Kernel 0: zero workspace accumulators
// ---------------------------------------------------------------------------
__global__ void egnn_zero_kernel(float* __restrict__ p, size_t n) {
    size_t i = (size_t)blockIdx.x * blockDim.x + threadIdx.x;
    if (i < n) p[i] = 0.0f;
}

// ---------------------------------------------------------------------------
// Kernel 1: fused per-edge  gather -> msgMLP -> posMLP -> atomic scatter
// One wave = 16 edges. Block = 256 threads = 8 waves = 128 edges.
// ---------------------------------------------------------------------------
__global__ __launch_bounds__(256)
void egnn_edge_kernel(const float* __restrict__ h, const float* __restrict__ pos,
                      const int* __restrict__ eidx,
                      const float* __restrict__ Wm1, const float* __restrict__ bm1,
                      const float* __restrict__ Wm2, const float* __restrict__ bm2,
                      const float* __restrict__ Wp1, const float* __restrict__ bp1,
                      const float* __restrict__ Wp2, const float* __restrict__ bp2,
                      float* __restrict__ aggM, float* __restrict__ aggP,
                      float* __restrict__ cnt, int E) {
    __shared__ float Wm1z[64 * 128];       // rows 0..127, K-pair swizzled (32 KB)
    __shared__ float Wm2z[32 * 128];       // 16 KB
    __shared__ float Wp1z[32 * 128];       // 16 KB
    __shared__ float stage[8][16 * DD];    // per-wave 16x64 re-layout buffer (32 KB)
    __shared__ float d2w[8][16];
    __shared__ float wbuf[8][16];
    __shared__ int   dstw[8][16];
    __shared__ int   valw[8][16];

    const int tid  = threadIdx.x;
    const int lane = tid & 31;
    const int wave = tid >> 5;
    const int ln   = lane & 15;
    const int half = lane >> 4;

    for (int i = tid; i < 128 * DD; i += 256) {
        const int k = i >> 6, n = i & 63;
        Wm1z[(k >> 1) * 128 + (n << 1) + (k & 1)] = Wm1[i];
    }
    for (int i = tid; i < 64 * DD; i += 256) {
        const int k = i >> 6, n = i & 63;
        Wm2z[(k >> 1) * 128 + (n << 1) + (k & 1)] = Wm2[i];
        Wp1z[(k >> 1) * 128 + (n << 1) + (k & 1)] = Wp1[i];
    }

    // ---- per-edge metadata (lane ln owns edge e0+ln; both halves duplicate) --
    const int e0    = (blockIdx.x * 8 + wave) * 16;
    const int eU    = e0 + ln;
    const int valid = (eU < E) ? 1 : 0;
    const int e     = valid ? eU : (E - 1);
    const int s     = eidx[e];          // source j
    const int d     = eidx[E + e];      // target i
    const float rx  = pos[2 * d]     - pos[2 * s];
    const float ry  = pos[2 * d + 1] - pos[2 * s + 1];
    const float d2  = rx * rx + ry * ry;
    if (half == 0) { d2w[wave][ln] = d2; dstw[wave][ln] = d; valw[wave][ln] = valid; }
    __syncthreads();

    // ======================= GEMM1: [16x129] @ Wm1 =========================
    // C init folds bias + d2-column (K=128) rank-1 term.
    float d2v[8];
#pragma unroll
    for (int r = 0; r < 8; ++r) d2v[r] = d2w[wave][r + 8 * half];

    v8f c1[4];
#pragma unroll
    for (int t = 0; t < 4; ++t) {
        const int n = t * 16 + ln;
        const float bb = bm1[n];
        const float wl = Wm1[128 * DD + n];       // d2 row, read once from L2
#pragma unroll
        for (int r = 0; r < 8; ++r) c1[t][r] = bb + d2v[r] * wl;
    }

    const size_t rowI = (size_t)d * DD;   // h_i = h[dst]
    const size_t rowJ = (size_t)s * DD;   // h_j = h[src]
#pragma unroll 4
    for (int k = 0; k < 128; k += 4) {
        const int kk = k + 2 * half;      // this lane's K pair (kk even)
        const float* xp = (kk < DD) ? (h + rowI + kk) : (h + rowJ + (kk - DD));
        const v2f a = *(const v2f*)xp;
        const float* bz = &Wm1z[((k >> 1) + half) * 128 + 2 * ln];
#pragma unroll
        for (int t = 0; t < 4; ++t)
            c1[t] = wmma4(a, *(const v2f*)(bz + 32 * t), c1[t]);
    }
    // ReLU + re-layout C->A via per-wave LDS stage
#pragma unroll
    for (int t = 0; t < 4; ++t)
#pragma unroll
        for (int r = 0; r < 8; ++r)
            stage[wave][(r + 8 * half) * DD + t * 16 + ln] = fmaxf(c1[t][r], 0.0f);
    __builtin_amdgcn_wave_barrier();

    // ======================= GEMM2: m1 @ Wm2 ===============================
    v8f c2[4];
#pragma unroll
    for (int t = 0; t < 4; ++t) {
        const float bb = bm2[t * 16 + ln];
#pragma unroll
        for (int r = 0; r < 8; ++r) c2[t][r] = bb;
    }
#pragma unroll 4
    for (int k = 0; k < DD; k += 4) {
        const int kk = k + 2 * half;
        const v2f a = *(const v2f*)&stage[wave][ln * DD + kk];
        const float* bz = &Wm2z[((k >> 1) + half) * 128 + 2 * ln];
#pragma unroll
        for (int t = 0; t < 4; ++t)
            c2[t] = wmma4(a, *(const v2f*)(bz + 32 * t), c2[t]);
    }
#pragma unroll
    for (int t = 0; t < 4; ++t)
#pragma unroll
        for (int r = 0; r < 8; ++r) c2[t][r] = fmaxf(c2[t][r], 0.0f);
    __builtin_amdgcn_wave_barrier();

    // stage m2 for GEMM3, and scatter-sum messages into aggM (L2-resident)
#pragma unroll
    for (int t = 0; t < 4; ++t)
#pragma unroll
        for (int r = 0; r < 8; ++r)
            stage[wave][(r + 8 * half) * DD + t * 16 + ln] = c2[t][r];
#pragma unroll
    for (int t = 0; t < 4; ++t)
#pragma unroll
        for (int r = 0; r < 8; ++r) {
            const int row = r + 8 * half;
            if (valw[wave][row])
                atomicAdd(aggM + (size_t)dstw[wave][row] * DD + t * 16 + ln, c2[t][r]);
        }
    __builtin_amdgcn_wave_barrier();

    // ======================= GEMM3: m2 @ Wp1 ===============================
    v8f c3[4];
#pragma unroll
    for (int t = 0; t < 4; ++t) {
        const float bb = bp1[t * 16 + ln];
#pragma unroll
        for (int r = 0; r < 8; ++r) c3[t][r] = bb;
    }
#pragma unroll 4
    for (int k = 0; k < DD; k += 4) {
        const int kk = k + 2 * half;
        const v2f a = *(const v2f*)&stage[wave][ln * DD + kk];
        const float* bz = &Wp1z[((k >> 1) + half) * 128 + 2 * ln];
#pragma unroll
        for (int t = 0; t < 4; ++t)
            c3[t] = wmma4(a, *(const v2f*)(bz + 32 * t), c3[t]);
    }
    // w[e] = ReLU(t) @ Wp2 + bp2 : cross-lane reduce inside 16-lane halves
    const float bp2v = bp2[0];
    float wp2v[4];
#pragma unroll
    for (int t = 0; t < 4; ++t) wp2v[t] = Wp2[t * 16 + ln];
    float part[8];
#pragma unroll
    for (int r = 0; r < 8; ++r) {
        float p = 0.0f;
#pragma unroll
        for (int t = 0; t < 4; ++t) p += fmaxf(c3[t][r], 0.0f) * wp2v[t];
        part[r] = p;
    }
#pragma unroll
    for (int m = 1; m <= 8; m <<= 1)
#pragma unroll
        for (int r = 0; r < 8; ++r) part[r] += __shfl_xor(part[r], m, 32);
    if (ln == 0) {
#pragma unroll
        for (int r = 0; r < 8; ++r) wbuf[wave][r + 8 * half] = part[r] + bp2v;
    }
    __builtin_amdgcn_wave_barrier();

    // pos-message scatter (+ degree count), one lane per edge
    if (half == 0 && valid) {
        const float wv = wbuf[wave][ln];
        atomicAdd(aggP + (size_t)d * 2 + 0, rx * wv);
        atomicAdd(aggP + (size_t)d * 2 + 1, ry * wv);
        atomicAdd(cnt + d, 1.0f);
    }
}

// ---------------------------------------------------------------------------
// Kernel 2: per-node update MLP + residual + pos mean-update
// ---------------------------------------------------------------------------
__global__ __launch_bounds__(256)
void egnn_node_kernel(const float* __restrict__ h, const float* __restrict__ pos,
                      const float* __restrict__ Wu1, const float* __restrict__ bu1,
                      const float* __restrict__ Wu2, const float* __restrict__ bu2,
                      const float* __restrict__ aggM, const float* __restrict__ aggP,
                      const float* __restrict__ cnt, float* __restrict__ out, int N) {
    __shared__ float Wu1z[64 * 128];       // 32 KB, K-pair swizzled
    __shared__ float Wu2z[32 * 128];       // 16 KB
    __shared__ float stage[8][16 * DD];    // 32 KB

    const int tid  = threadIdx.x;
    const int lane = tid & 31;
    const int wave = tid >> 5;
    const int ln   = lane & 15;
    const int half = lane >> 4;

    for (int i = tid; i < 128 * DD; i += 256) {
        const int k = i >> 6, n = i & 63;
        Wu1z[(k >> 1) * 128 + (n << 1) + (k & 1)] = Wu1[i];
    }
    for (int i = tid; i < 64 * DD; i += 256) {
        const int k = i >> 6, n = i & 63;
        Wu2z[(k >> 1) * 128 + (n << 1) + (k & 1)] = Wu2[i];
    }
    __syncthreads();

    const int n0   = (blockIdx.x * 8 + wave) * 16;
    const int myU  = n0 + ln;
    const int node = (myU < N) ? myU : (N - 1);

    // ============ GEMM1: [h | aggM] (16x128) @ Wu1 =========================
    v8f c1[4];
#pragma unroll
    for (int t = 0; t < 4; ++t) {
        const float bb = bu1[t * 16 + ln];
#pragma unroll
        for (int r = 0; r < 8; ++r) c1[t][r] = bb;
    }
    const size_t rowH = (size_t)node * DD;
#pragma unroll 4
    for (int k = 0; k < 128; k += 4) {
        const int kk = k + 2 * half;
        const float* xp = (kk < DD) ? (h + rowH + kk) : (aggM + rowH + (kk - DD));
        const v2f a = *(const v2f*)xp;
        const float* bz = &Wu1z[((k >> 1) + half) * 128 + 2 * ln];
#pragma unroll
        for (int t = 0; t < 4; ++t)
            c1[t] = wmma4(a, *(const v2f*)(bz + 32 * t), c1[t]);
    }
#pragma unroll
    for (int t = 0; t < 4; ++t)
#pragma unroll
        for (int r = 0; r < 8; ++r)
            stage[wave][(r + 8 * half) * DD + t * 16 + ln] = fmaxf(c1[t][r], 0.0f);
    __builtin_amdgcn_wave_barrier();

    // ============ GEMM2: @ Wu2 + residual ==================================
    v8f c2[4];
#pragma unroll
    for (int t = 0; t < 4; ++t) {
        const float bb = bu2[t * 16 + ln];
#pragma unroll
        for (int r = 0; r < 8; ++r) c2[t][r] = bb;
    }
#pragma unroll 4
    for (int k = 0; k < DD; k += 4) {
        const int kk = k + 2 * half;
        const v2f a = *(const v2f*)&stage[wave][ln * DD + kk];
        const float* bz = &Wu2z[((k >> 1) + half) * 128 + 2 * ln];
#pragma unroll
        for (int t = 0; t < 4; ++t)
            c2[t] = wmma4(a, *(const v2f*)(bz + 32 * t), c2[t]);
    }
#pragma unroll
    for (int t = 0; t < 4; ++t)
#pragma unroll
        for (int r = 0; r < 8; ++r) {
            const int row = r + 8 * half;
            const int nd  = n0 + row;
            if (nd < N) {
                const int col = t * 16 + ln;
                out[(size_t)nd * DD + col] = h[(size_t)nd * DD + col] + c2[t][r];
            }
        }

    // pos_out = pos + aggP / max(cnt, 1)
    if (half == 0 && myU < N) {
        const float cc = fmaxf(cnt[myU], 1.0f);
        float* po = out + (size_t)N * DD;
        po[(size_t)myU * 2 + 0] = pos[(size_t)myU * 2 + 0] + aggP[(size_t)myU * 2 + 0] / cc;
        po[(size_t)myU * 2 + 1] = pos[(size_t)myU * 2 + 1] + aggP[(size_t)myU * 2 + 1] / cc;
    }
}

// ---------------------------------------------------------------------------
extern "C" void kernel_launch(void* const* d_in, const int* in_sizes, int n_in,
                              void* d_out, int out_size, void* d_ws, size_t ws_size,
                              hipStream_t stream) {
    const float* h    = (const float*)d_in[0];
    const float* pos  = (const float*)d_in[1];
    const int*   eidx = (const int*)d_in[2];
    // d_in[3] = h_init (unused: reinit=False)
    const float* Wm1 = (const float*)d_in[4];  const float* bm1 = (const float*)d_in[5];
    const float* Wm2 = (const float*)d_in[6];  const float* bm2 = (const float*)d_in[7];
    const float* Wp1 = (const float*)d_in[8];  const float* bp1 = (const float*)d_in[9];
    const float* Wp2 = (const float*)d_in[10]; const float* bp2 = (const float*)d_in[11];
    const float* Wu1 = (const float*)d_in[12]; const float* bu1 = (const float*)d_in[13];
    const float* Wu2 = (const float*)d_in[14]; const float* bu2 = (const float*)d_in[15];

    const int N = in_sizes[0] / DD;
    const int E = in_sizes[2] / 2;

    float* aggM = (float*)d_ws;                 // [N,64]
    float* aggP = aggM + (size_t)N * DD;        // [N,2]
    float* cntv = aggP + (size_t)N * 2;         // [N]

    const size_t nz = (size_t)N * (DD + 3);
    egnn_zero_kernel<<<(unsigned)((nz + 255) / 256), 256, 0, stream>>>(aggM, nz);
    egnn_edge_kernel<<<(E + 127) / 128, 256, 0, stream>>>(
        h, pos, eidx, Wm1, bm1, Wm2, bm2, Wp1, bp1, Wp2, bp2, aggM, aggP, cntv, E);
    egnn_node_kernel<<<(N + 127) / 128, 256, 0, stream>>>(
        h, pos, Wu1, bu1, Wu2, bu2, aggM, aggP, cntv, (float*)d_out, N);
}